// SparseLayer_54142357733912
// MI455X (gfx1250) — compile-verified
//
#include <hip/hip_runtime.h>
#include <stdint.h>

#define N_NEURONS 50000
#define N_INPUT   17400
#define NNZ_TOT   1000000
#define T_STEPS   256
#define TILE_N    32      // rows per SpMM workgroup
#define LDS_PITCH 36      // padded pitch (floats) -> low LDS bank conflicts
#define STAGE_CAP 1536    // max (col,w) pairs staged in LDS per tile (12 KB)

#define GAS __attribute__((address_space(1)))
#define LAS __attribute__((address_space(3)))

#if __has_builtin(__builtin_amdgcn_global_store_async_from_lds_b32)
#define HAVE_ASYNC_STORE 1
#endif
#if __has_builtin(__builtin_amdgcn_global_load_async_to_lds_b32)
#define HAVE_ASYNC_LOAD 1
#endif

#if defined(HAVE_ASYNC_STORE) || defined(HAVE_ASYNC_LOAD)
__device__ __forceinline__ void wait_async_zero() {
#if __has_builtin(__builtin_amdgcn_s_wait_asynccnt)
  __builtin_amdgcn_s_wait_asynccnt(0);
#else
  asm volatile("s_wait_asynccnt 0" ::: "memory");
#endif
}
#endif

// ---------------------------------------------------------------------------
// 1) Transpose x: (T=256, N_INPUT) row-major -> xT: (N_INPUT, T) row-major.
//    Makes every sparse column gather in the SpMM a contiguous 1KB read.
// ---------------------------------------------------------------------------
__global__ __launch_bounds__(256) void transpose_kernel(
    const float* __restrict__ x, float* __restrict__ xT) {
  __shared__ float tile[32][33];
  const int c0 = blockIdx.x * 32;
  const int t0 = blockIdx.y * 32;
  const int lx = threadIdx.x;          // 0..31
  const int ly = threadIdx.y;          // 0..7
  for (int i = 0; i < 32; i += 8) {
    const int c = c0 + lx;
    const int t = t0 + ly + i;         // always < 256
    tile[ly + i][lx] = (c < N_INPUT) ? x[t * N_INPUT + c] : 0.0f;
  }
  __syncthreads();
  for (int i = 0; i < 32; i += 8) {
    const int c = c0 + ly + i;
    const int t = t0 + lx;
    if (c < N_INPUT) xT[c * T_STEPS + t] = tile[lx][ly + i];
  }
}

// ---------------------------------------------------------------------------
// 2) Row histogram (COO -> per-row nnz counts).
// ---------------------------------------------------------------------------
__global__ __launch_bounds__(256) void hist_kernel(
    const int* __restrict__ rows, int* __restrict__ counts) {
  const int k = blockIdx.x * blockDim.x + threadIdx.x;
  if (k < NNZ_TOT) atomicAdd(&counts[rows[k]], 1);
}

// ---------------------------------------------------------------------------
// 3) Exclusive prefix scan over 50000 counts (single 1024-thread workgroup,
//    Hillis-Steele per chunk + running carry). One-shot, tiny cost.
// ---------------------------------------------------------------------------
__global__ __launch_bounds__(1024) void scan_kernel(
    const int* __restrict__ counts, int* __restrict__ offsets,
    int* __restrict__ cursor) {
  __shared__ int s[1024];
  const int tid = threadIdx.x;
  int carry = 0;
  for (int base = 0; base < N_NEURONS; base += 1024) {
    const int i = base + tid;
    const int v = (i < N_NEURONS) ? counts[i] : 0;
    s[tid] = v;
    __syncthreads();
    for (int off = 1; off < 1024; off <<= 1) {
      const int addend = (tid >= off) ? s[tid - off] : 0;
      __syncthreads();
      s[tid] += addend;
      __syncthreads();
    }
    const int excl = s[tid] - v;
    if (i < N_NEURONS) {
      offsets[i] = carry + excl;
      cursor[i]  = carry + excl;
    }
    const int total = s[1023];
    __syncthreads();               // protect s[] before next chunk overwrites
    carry += total;
  }
  if (tid == 0) offsets[N_NEURONS] = NNZ_TOT;
}

// ---------------------------------------------------------------------------
// 4) Scatter COO entries into CSR slots, packing (col, weight-bits) into one
//    int2 so the SpMM hot loop does a single uniform 8B load per nonzero.
// ---------------------------------------------------------------------------
__global__ __launch_bounds__(256) void scatter_kernel(
    const int* __restrict__ rows, const int* __restrict__ cols,
    const float* __restrict__ w, int* __restrict__ cursor,
    int2* __restrict__ pairs) {
  const int k = blockIdx.x * blockDim.x + threadIdx.x;
  if (k < NNZ_TOT) {
    const int n = rows[k];
    const int p = atomicAdd(&cursor[n], 1);
    pairs[p] = make_int2(cols[k], __float_as_int(w[k]));
  }
}

// ---------------------------------------------------------------------------
// 5) CSR SpMM + background term. Block = 256 threads (8 wave32) handles
//    TILE_N=32 consecutive rows x all 256 timesteps.
//    - Tile's CSR pairs (contiguous range) are DMA'd into LDS with
//      global_load_async_to_lds_b32 (ASYNCcnt path); hot loop then reads them
//      as broadcast ds_load_b64, leaving VMEM entirely to xT gathers.
//    - Thread == timestep in the gather phase: coalesced 128B xT reads,
//      all L2-resident (xT = 17.8 MB << 192 MB L2).
//    - Results transposed through padded LDS and streamed out with coalesced
//      async-from-LDS stores.
// ---------------------------------------------------------------------------
__global__ __launch_bounds__(256) void spmm_kernel(
    const float* __restrict__ xT, const int2* __restrict__ pairs,
    const int* __restrict__ offsets, const float* __restrict__ bkg,
    float* __restrict__ out) {
  __shared__ float tile[T_STEPS * LDS_PITCH];   // 36,864 B
  __shared__ int2  stage[STAGE_CAP];            // 12,288 B
  const int t  = threadIdx.x;                   // timestep 0..255
  const int n0 = blockIdx.x * TILE_N;
  const int n_end = (n0 + TILE_N < N_NEURONS) ? (n0 + TILE_N) : N_NEURONS;

  const int base     = offsets[n0];
  const int tile_end = offsets[n_end];
  const int tile_nnz = tile_end - base;
  const bool staged  = (tile_nnz <= STAGE_CAP); // uniform across block

  if (staged) {
    const int  ndw = tile_nnz * 2;              // dwords to copy
    const int* gi  = (const int*)(pairs + base);
    int*       li  = (int*)stage;
    for (int i = t; i < ndw; i += 256) {
#ifdef HAVE_ASYNC_LOAD
      __builtin_amdgcn_global_load_async_to_lds_b32(
          (GAS int*)(uintptr_t)(gi + i),
          (LAS int*)(uint32_t)(uintptr_t)(li + i), 0, 0);
#else
      li[i] = gi[i];
#endif
    }
#ifdef HAVE_ASYNC_LOAD
    wait_async_zero();
#endif
  } else {
    __builtin_prefetch(pairs + base);           // global_prefetch_b8
  }
  __syncthreads();                              // staged data visible to all waves

  for (int r = 0; r < TILE_N; ++r) {
    const int n = n0 + r;
    float acc = 0.0f;
    if (n < N_NEURONS) {
      const int beg = offsets[n];
      const int end = offsets[n + 1];
      acc = 4.0f * bkg[n];                      // rest == BKG_SPIKE_RATE == 4
      if (staged) {
        for (int j = beg; j < end; ++j) {
          const int2 e = stage[j - base];       // broadcast ds_load_b64
          acc += __int_as_float(e.y) * xT[e.x * T_STEPS + t];
        }
      } else {
        for (int j = beg; j < end; ++j) {
          const int2 e = pairs[j];              // uniform global b64
          acc += __int_as_float(e.y) * xT[e.x * T_STEPS + t];
        }
      }
    }
    tile[t * LDS_PITCH + r] = acc;              // (t-major, row contiguous)
  }
  __syncthreads();

  // Write-out: lanes cover 32 consecutive neurons at a fixed timestep ->
  // 128B contiguous global stores per wave, issued async from LDS.
  const int nl = t & (TILE_N - 1);
  const int tb = t >> 5;                        // 0..7
  const int n  = n0 + nl;
  if (n < N_NEURONS) {
    for (int i = 0; i < TILE_N; ++i) {
      const int tt = tb + i * 8;
      float* gp = out + (size_t)tt * N_NEURONS + n;
      const float* lp = &tile[tt * LDS_PITCH + nl];
#ifdef HAVE_ASYNC_STORE
      __builtin_amdgcn_global_store_async_from_lds_b32(
          (GAS int*)(uintptr_t)gp,
          (LAS int*)(uint32_t)(uintptr_t)lp, 0, 0);
#else
      *gp = *lp;
#endif
    }
  }
#ifdef HAVE_ASYNC_STORE
  wait_async_zero();
#endif
}

// ---------------------------------------------------------------------------
// Host launcher
// ---------------------------------------------------------------------------
static inline size_t align_up256(size_t v) { return (v + 255) & ~(size_t)255; }

extern "C" void kernel_launch(void* const* d_in, const int* in_sizes, int n_in,
                              void* d_out, int out_size, void* d_ws, size_t ws_size,
                              hipStream_t stream) {
  (void)in_sizes; (void)n_in; (void)out_size; (void)ws_size;

  const float* inp     = (const float*)d_in[0];  // (1, 256, 17400)
  const float* weights = (const float*)d_in[1];  // (1e6,)
  const float* bkg     = (const float*)d_in[2];  // (50000,)
  const int*   rows    = (const int*)d_in[3];    // (1e6,)
  const int*   cols    = (const int*)d_in[4];    // (1e6,)
  float*       out     = (float*)d_out;          // (1, 256, 50000)

  char* ws = (char*)d_ws;
  size_t off = 0;
  float* xT      = (float*)(ws + off); off = align_up256(off + sizeof(float) * N_INPUT * T_STEPS);
  int*   counts  = (int*)  (ws + off); off = align_up256(off + sizeof(int)   * N_NEURONS);
  int*   offsets = (int*)  (ws + off); off = align_up256(off + sizeof(int)   * (N_NEURONS + 1));
  int*   cursor  = (int*)  (ws + off); off = align_up256(off + sizeof(int)   * N_NEURONS);
  int2*  pairs   = (int2*) (ws + off); off = align_up256(off + sizeof(int2)  * NNZ_TOT);
  // total workspace use: ~26.5 MB

  (void)hipMemsetAsync(counts, 0, sizeof(int) * N_NEURONS, stream);

  transpose_kernel<<<dim3((N_INPUT + 31) / 32, T_STEPS / 32), dim3(32, 8), 0, stream>>>(inp, xT);
  hist_kernel<<<(NNZ_TOT + 255) / 256, 256, 0, stream>>>(rows, counts);
  scan_kernel<<<1, 1024, 0, stream>>>(counts, offsets, cursor);
  scatter_kernel<<<(NNZ_TOT + 255) / 256, 256, 0, stream>>>(rows, cols, weights, cursor, pairs);
  spmm_kernel<<<(N_NEURONS + TILE_N - 1) / TILE_N, 256, 0, stream>>>(xT, pairs, offsets, bkg, out);
}